// loss_86681029968066
// MI455X (gfx1250) — compile-verified
//
#include <hip/hip_runtime.h>

typedef float v2f __attribute__((ext_vector_type(2)));
typedef float v8f __attribute__((ext_vector_type(8)));

#define B_   64
#define H_   512
#define W_   512
#define EPSF 1e-5f

#define WAVES 8      // waves per block, one 16x16 output tile each
#define KEXT  24     // 16 output rows + 8 halo rows
#define CEXT  144    // 128 output cols + 16 halo/slop cols
#define TCOLS 24     // T columns actually needed per wave (16 + 8 halo)

// ---------------------------------------------------------------------------
// ws layout (doubles): [0]=sum(cc), [1]=sum(dx^2), [2]=sum(dy^2), [3]=spare
// ---------------------------------------------------------------------------

__global__ void init_kernel(double* ws) {
  if (threadIdx.x < 4) ws[threadIdx.x] = 0.0;
}

// ---------------------------------------------------------------------------
// NCC kernel: fused 9x9 box sums (vertical stage via WMMA band-matmul,
// horizontal stage via LDS sliding sums) + cc reduction.
// ---------------------------------------------------------------------------
__global__ __launch_bounds__(256) void ncc_kernel(const float* __restrict__ I,
                                                  const float* __restrict__ J,
                                                  double* __restrict__ acc) {
  __shared__ float Is[KEXT][CEXT];
  __shared__ float Js[KEXT][CEXT];
  __shared__ float Ts[WAVES][5][16][TCOLS];
  __shared__ float red[256];

  const int tid  = threadIdx.x;
  const int bid  = blockIdx.x;
  const int tcg  = bid & 3;          // which 128-col strip
  const int trow = (bid >> 2) & 31;  // which 16-row strip
  const int b    = bid >> 7;         // batch image
  const int r0   = trow * 16;
  const int c0   = tcg * 128;

  const float* Ib = I + (size_t)b * (H_ * W_);
  const float* Jb = J + (size_t)b * (H_ * W_);

  // --- stage 0: load halo'd input region (zero padding = SAME conv pad) ---
  for (int idx = tid; idx < KEXT * CEXT; idx += 256) {
    const int rr = idx / CEXT, cc = idx % CEXT;
    const int gr = r0 - 4 + rr, gc = c0 - 4 + cc;
    float vi = 0.f, vj = 0.f;
    if (gr >= 0 && gr < H_ && gc >= 0 && gc < W_) {
      vi = Ib[gr * W_ + gc];
      vj = Jb[gr * W_ + gc];
    }
    Is[rr][cc] = vi;
    Js[rr][cc] = vj;
  }
  __syncthreads();

  const int w    = tid >> 5;   // wave id = tile id within strip
  const int lane = tid & 31;
  const int m    = lane & 15;  // A-matrix row / B-matrix column index
  const int half = lane >> 4;

  v8f zero = {};
  v8f accq[5][2];
  #pragma unroll
  for (int q = 0; q < 5; ++q) {
    accq[q][0] = zero;
    accq[q][1] = zero;
  }

  // --- stage 1: vertical 9-row window sums as band matmul on WMMA pipe ---
  // T[m][n] = sum_{k=m..m+8} X[k][n], X rows = input rows r0-4 .. r0+19
  #pragma unroll
  for (int k0 = 0; k0 < 24; k0 += 4) {
    const int kA0 = k0 + 2 * half;  // K index held in vgpr0 for this lane
    const int kA1 = kA0 + 1;        // K index held in vgpr1
    v2f a;
    a.x = (kA0 >= m && kA0 <= m + 8) ? 1.f : 0.f;
    a.y = (kA1 >= m && kA1 <= m + 8) ? 1.f : 0.f;
    #pragma unroll
    for (int g = 0; g < 2; ++g) {
      const int col = 16 * w + 16 * g + m;  // LDS col (global col c0-4+...)
      const float i0 = Is[kA0][col], i1 = Is[kA1][col];
      const float j0 = Js[kA0][col], j1 = Js[kA1][col];
      v2f bI  = {i0, i1};
      v2f bJ  = {j0, j1};
      v2f bI2 = {i0 * i0, i1 * i1};
      v2f bJ2 = {j0 * j0, j1 * j1};
      v2f bIJ = {i0 * j0, i1 * j1};
      accq[0][g] = __builtin_amdgcn_wmma_f32_16x16x4_f32(false, a, false, bI,
                     (short)0, accq[0][g], false, false);
      accq[1][g] = __builtin_amdgcn_wmma_f32_16x16x4_f32(false, a, false, bJ,
                     (short)0, accq[1][g], false, false);
      accq[2][g] = __builtin_amdgcn_wmma_f32_16x16x4_f32(false, a, false, bI2,
                     (short)0, accq[2][g], false, false);
      accq[3][g] = __builtin_amdgcn_wmma_f32_16x16x4_f32(false, a, false, bJ2,
                     (short)0, accq[3][g], false, false);
      accq[4][g] = __builtin_amdgcn_wmma_f32_16x16x4_f32(false, a, false, bIJ,
                     (short)0, accq[4][g], false, false);
    }
  }

  // --- spill T to LDS using documented 16x16 f32 C/D layout ---
  // VGPR r: lanes 0-15 -> (M=r, N=lane), lanes 16-31 -> (M=r+8, N=lane-16)
  #pragma unroll
  for (int q = 0; q < 5; ++q) {
    #pragma unroll
    for (int g = 0; g < 2; ++g) {
      const int N = 16 * g + m;
      if (N < TCOLS) {
        #pragma unroll
        for (int r = 0; r < 8; ++r) {
          const int M = r + 8 * half;
          Ts[w][q][M][N] = accq[q][g][r];
        }
      }
    }
  }

  // --- stage 2: horizontal 9-col window sums + NCC per pixel ---
  float partial = 0.f;
  const int nOut = m;  // output column within tile handled by this lane
  #pragma unroll
  for (int jr = 0; jr < 8; ++jr) {
    const int M = 8 * half + jr;  // output row within tile
    float S[5];
    #pragma unroll
    for (int q = 0; q < 5; ++q) {
      float s = 0.f;
      #pragma unroll
      for (int t = 0; t < 9; ++t) s += Ts[w][q][M][nOut + t];
      S[q] = s;
    }
    const float inv_win = 1.0f / 81.0f;
    const float cross = S[4] - S[0] * S[1] * inv_win;
    const float Ivar  = S[2] - S[0] * S[0] * inv_win;
    const float Jvar  = S[3] - S[1] * S[1] * inv_win;
    partial += cross * cross / (Ivar * Jvar + EPSF);
  }

  // --- block reduction + f64 atomic accumulate ---
  red[tid] = partial;
  __syncthreads();
  for (int s = 128; s > 0; s >>= 1) {
    if (tid < s) red[tid] += red[tid + s];
    __syncthreads();
  }
  if (tid == 0) atomicAdd(acc, (double)red[0]);
}

// ---------------------------------------------------------------------------
// Regularizer: sum of squared forward differences of phi along x and y.
// ---------------------------------------------------------------------------
__global__ __launch_bounds__(256) void reg_kernel(const float* __restrict__ P,
                                                  double* __restrict__ accdx,
                                                  double* __restrict__ accdy) {
  __shared__ float rx[256];
  __shared__ float ry[256];
  const long long N = (long long)B_ * 2 * H_ * W_;
  float sx = 0.f, sy = 0.f;
  for (long long i = (long long)blockIdx.x * blockDim.x + threadIdx.x; i < N;
       i += (long long)gridDim.x * blockDim.x) {
    const int col = (int)(i % W_);
    const int row = (int)((i / W_) % H_);
    const float v = P[i];
    if (col < W_ - 1) { const float d = P[i + 1] - v;  sx += d * d; }
    if (row < H_ - 1) { const float d = P[i + W_] - v; sy += d * d; }
  }
  const int tid = threadIdx.x;
  rx[tid] = sx;
  ry[tid] = sy;
  __syncthreads();
  for (int s = 128; s > 0; s >>= 1) {
    if (tid < s) { rx[tid] += rx[tid + s]; ry[tid] += ry[tid + s]; }
    __syncthreads();
  }
  if (tid == 0) {
    atomicAdd(accdx, (double)rx[0]);
    atomicAdd(accdy, (double)ry[0]);
  }
}

// ---------------------------------------------------------------------------
// Finalize: 3 scalar outputs (loss_sim, loss_reg, total)
// ---------------------------------------------------------------------------
__global__ void finalize_kernel(const double* __restrict__ ws,
                                float* __restrict__ out) {
  const double npix = (double)B_ * H_ * W_;
  const double ndx  = (double)B_ * 2.0 * H_ * (W_ - 1);
  const double ndy  = (double)B_ * 2.0 * (H_ - 1) * W_;
  const double loss_sim = -(ws[0] / npix);                       // SIM_FACTOR=1
  const double loss_reg = 0.1 * 0.5 * (ws[1] / ndx + ws[2] / ndy);
  out[0] = (float)loss_sim;
  out[1] = (float)loss_reg;
  out[2] = (float)(loss_sim + loss_reg);
}

extern "C" void kernel_launch(void* const* d_in, const int* in_sizes, int n_in,
                              void* d_out, int out_size, void* d_ws, size_t ws_size,
                              hipStream_t stream) {
  (void)in_sizes; (void)n_in; (void)out_size; (void)ws_size;
  const float* I = (const float*)d_in[0];   // target_proj
  const float* J = (const float*)d_in[1];   // warped_moving
  const float* P = (const float*)d_in[2];   // phi
  double* ws = (double*)d_ws;
  float* out = (float*)d_out;

  init_kernel<<<1, 4, 0, stream>>>(ws);
  ncc_kernel<<<B_ * (H_ / 16) * (W_ / 128), 256, 0, stream>>>(I, J, ws + 0);
  reg_kernel<<<2048, 256, 0, stream>>>(P, ws + 1, ws + 2);
  finalize_kernel<<<1, 1, 0, stream>>>(ws, out);
}